// BakedAttentionHead_62380105007586
// MI455X (gfx1250) — compile-verified
//
#include <hip/hip_runtime.h>
#include <math.h>

// ---------------------------------------------------------------------------
// Fused "softmax1" attention for MI455X (gfx1250, wave32, WMMA + async DMA).
//   scores = (Q K^T) * D^-1/2 ; softmax with +1 denominator; reciprocal via
//   16-step sigmoid long division; out = weights @ V.
// MI455X reasoning: K/V are 8MB each -> L2-resident (192MB); Q/O stream once
// from HBM. Compute-bound (AI ~860 FLOP/B) on the fp32 tensor pipes
// (v_wmma_f32_16x16x4_f32; fp32 kept because scores feed exp()).
// K tiles DMA'd with GLOBAL_LOAD_ASYNC_TO_LDS_B128 into per-wave
// double-buffered LDS panels (barrier-free score GEMM, ASYNCcnt-pipelined).
// Each wave owns 16 keys of a 128-key tile and computes a 32x16 S tile
// (two wmmas share each B fragment -> 3 LDS dwords per wmma). Softmax is
// parallel: 8 lanes per row + __shfl_xor reductions.
// ---------------------------------------------------------------------------

typedef float v2f __attribute__((ext_vector_type(2)));
typedef float v8f __attribute__((ext_vector_type(8)));

#define DDIM   1024
#define BM     32            // query rows per workgroup
#define BN     128           // keys per N-tile (16 per wave)
#define DC     128           // K-chunk depth per DMA batch
#define QSTR   1036          // Q row stride: 16B-aligned rows, banks distinct
#define KSTR   132           // K panel row stride: 16B-aligned, banks distinct
#define SSTR   129           // S row stride: banks distinct (129 % 64 = 1)
#define SIGK   100.0f        // sigmoid sharpness of long-division comparator
#define PBITS  16

// LDS layout (floats)
#define KPAN   (16 * KSTR)                 // one 16-row K panel = 2112 floats
#define OFF_Q  0
#define OFF_K  (OFF_Q + BM * QSTR)         // 33152
#define OFF_S  (OFF_K + 8 * 2 * KPAN)      // +33792 = 66944
#define OFF_M  (OFF_S + BM * SSTR)         // +4128  = 71072
#define OFF_L  (OFF_M + BM)
#define OFF_A  (OFF_L + BM)
#define OFF_R  (OFF_A + BM)
#define SMEM_FLOATS (OFF_R + BM)           // 71200 floats = 284800 bytes

// ---- CDNA5 async DMA helpers (inline asm; portable across toolchains) ------
__device__ __forceinline__ void async_b128(unsigned int lds_off, const float* g) {
    asm volatile("global_load_async_to_lds_b128 %0, %1, off"
                 :: "v"(lds_off), "v"(g) : "memory");
}
__device__ __forceinline__ unsigned int lds_off32(const void* p) {
    // flat LDS addresses carry the LDS byte offset in addr[31:0]
    return (unsigned int)(unsigned long long)p;
}
// issue one 16-row x DC-col K panel: 16 async B128 ops per wave (ASYNCcnt +16)
__device__ __forceinline__ void issue_k_panel(unsigned int lds_base,
                                              const float* gbase, int lane) {
    #pragma unroll
    for (int r = 0; r < 16; ++r) {
        async_b128(lds_base + (unsigned int)(r * KSTR + lane * 4) * 4u,
                   gbase + (size_t)r * DDIM + lane * 4);
    }
}
// one DC-deep slab of the 32x16 score tile: 64 x v_wmma_f32_16x16x4_f32,
// each B fragment shared by two A fragments (3 LDS dwords per wmma)
__device__ __forceinline__ void score_chunk2(v8f& s0, v8f& s1,
                                             const float* qrow0,
                                             const float* qrow1,
                                             const float* krow) {
    for (int kk = 0; kk < DC; kk += 4) {
        v2f a0, a1, b;
        a0.x = qrow0[kk + 0];
        a0.y = qrow0[kk + 1];
        a1.x = qrow1[kk + 0];
        a1.y = qrow1[kk + 1];
        b.x  = krow[kk + 0];
        b.y  = krow[kk + 1];
        s0 = __builtin_amdgcn_wmma_f32_16x16x4_f32(
            false, a0, false, b, (short)0, s0, false, false);
        s1 = __builtin_amdgcn_wmma_f32_16x16x4_f32(
            false, a1, false, b, (short)0, s1, false, false);
    }
}

extern "C" __global__ __launch_bounds__(256, 1)
void baked_attn_f32_wmma(const float* __restrict__ Qg,
                         const float* __restrict__ Kg,
                         const float* __restrict__ Vg,
                         float* __restrict__ Og,
                         int nkeys)
{
    extern __shared__ float smem[];
    float* Qs   = smem + OFF_Q;
    float* Ss   = smem + OFF_S;
    float* rowM = smem + OFF_M;
    float* rowL = smem + OFF_L;
    float* alph = smem + OFF_A;
    float* rrec = smem + OFF_R;

    const int tid  = threadIdx.x;
    const int lane = tid & 31;
    const int wave = tid >> 5;        // 8 waves
    const int lm   = lane & 15;       // row/col within 16
    const int lg   = lane >> 4;       // K-pair selector per ISA A/B layout
    const int mbase = blockIdx.x * BM;
    const float qscale = 0.03125f;    // 1024^-0.5 (folded into S; Q DMA'd raw)

    // ---- DMA Q tile into LDS: thread t covers column group t, rows 0..31 ---
    {
        const unsigned int qs_off = lds_off32(Qs);
        const float* qg = Qg + (size_t)mbase * DDIM + tid * 4;
        #pragma unroll 4
        for (int r = 0; r < BM; ++r) {
            async_b128(qs_off + (unsigned int)(r * QSTR + tid * 4) * 4u,
                       qg + (size_t)r * DDIM);
        }
    }
    if (tid < BM) { rowM[tid] = -__builtin_inff(); rowL[tid] = 0.0f; }
    asm volatile("s_wait_asynccnt 0x0" ::: "memory");
    __syncthreads();

    // output accumulator: 2 M-tiles x 8 col-tiles of 16 (wave owns 128 cols)
    v8f acc[2][8];
    const v8f vzero = {0.f, 0.f, 0.f, 0.f, 0.f, 0.f, 0.f, 0.f};
    #pragma unroll
    for (int i = 0; i < 2; ++i)
        #pragma unroll
        for (int j = 0; j < 8; ++j) acc[i][j] = vzero;

    const int c0 = wave * 128;   // this wave's output-column base

    // per-wave private double-buffered K panels (direct GEPs off smem: AS3)
    float* kbuf0 = smem + OFF_K + wave * 2 * KPAN;
    float* kbuf1 = kbuf0 + KPAN;
    const unsigned int koff0 = lds_off32(kbuf0);
    const unsigned int koff1 = lds_off32(kbuf1);

    for (int nb = 0; nb < nkeys; nb += BN) {
        // ------------- phase A: S = Q K^T over full D (barrier-free) -------
        // wave w owns keys [nb + 16w, nb + 16w + 16) and a 32x16 S tile
        const float* kg = Kg + (size_t)(nb + wave * 16) * DDIM;
        v8f sfrag0 = vzero, sfrag1 = vzero;

        asm volatile("s_wait_dscnt 0x0" ::: "memory"); // DS vs DMA unordered
        issue_k_panel(koff0, kg, lane);

        const float* qrow0 = Qs + (lm)      * QSTR + 2 * lg;
        const float* qrow1 = Qs + (16 + lm) * QSTR + 2 * lg;
        const float* krow0 = kbuf0 + lm * KSTR + 2 * lg;
        const float* krow1 = kbuf1 + lm * KSTR + 2 * lg;

        #pragma unroll 1
        for (int dc = 0; dc < DDIM; dc += 2 * DC) {
            // ---- chunk dc in buf0; prefetch dc+DC into buf1 ----
            asm volatile("s_wait_dscnt 0x0" ::: "memory");
            issue_k_panel(koff1, kg + dc + DC, lane);
            asm volatile("s_wait_asynccnt 0x10" ::: "memory");   // buf0 ready
            score_chunk2(sfrag0, sfrag1, qrow0 + dc, qrow1 + dc, krow0);

            // ---- chunk dc+DC in buf1; prefetch dc+2*DC into buf0 ----
            if (dc + 2 * DC < DDIM) {
                asm volatile("s_wait_dscnt 0x0" ::: "memory");
                issue_k_panel(koff0, kg + dc + 2 * DC, lane);
                asm volatile("s_wait_asynccnt 0x10" ::: "memory"); // buf1 ready
            } else {
                asm volatile("s_wait_asynccnt 0x0" ::: "memory");
            }
            score_chunk2(sfrag0, sfrag1, qrow0 + dc + DC, qrow1 + dc + DC, krow1);
        }

        // scatter wave's 32x16 S tile (apply D^-1/2); C-layout M = v + 8*lg
        #pragma unroll
        for (int v = 0; v < 8; ++v) {
            int m = v + 8 * lg;
            Ss[(m)      * SSTR + wave * 16 + lm] = sfrag0[v] * qscale;
            Ss[(16 + m) * SSTR + wave * 16 + lm] = sfrag1[v] * qscale;
        }
        __syncthreads();

        // ------- online softmax: 8 lanes per row + shfl reductions ---------
        {
            const int r  = tid >> 3;          // row 0..31 (8 lanes per row,
            const int sb = tid & 7;           //  all in one wave)
            float* s = Ss + r * SSTR + sb * 16;
            float mo = rowM[r];
            float tm = -__builtin_inff();
            #pragma unroll
            for (int j = 0; j < 16; ++j) tm = fmaxf(tm, s[j]);
            tm = fmaxf(tm, __shfl_xor(tm, 1, 8));
            tm = fmaxf(tm, __shfl_xor(tm, 2, 8));
            tm = fmaxf(tm, __shfl_xor(tm, 4, 8));
            float mn = fmaxf(mo, tm);
            float al = __expf(mo - mn);       // 0 on first tile (mo = -inf)
            float ps = 0.0f;
            #pragma unroll
            for (int j = 0; j < 16; ++j) {
                float e = __expf(s[j] - mn);
                s[j] = e;                     // Ss now holds P
                ps += e;
            }
            ps += __shfl_xor(ps, 1, 8);
            ps += __shfl_xor(ps, 2, 8);
            ps += __shfl_xor(ps, 4, 8);
            if (sb == 0) {
                rowM[r] = mn;
                rowL[r] = rowL[r] * al + ps;
                alph[r] = al;
            }
        }
        __syncthreads();

        // ------------- rescale accumulator by alpha[row] -------------------
        #pragma unroll
        for (int mt = 0; mt < 2; ++mt)
            #pragma unroll
            for (int v = 0; v < 8; ++v) {
                float al = alph[mt * 16 + v + 8 * lg];
                #pragma unroll
                for (int nt = 0; nt < 8; ++nt) acc[mt][nt][v] *= al;
            }

        // prefetch next key tile (global_prefetch_b8)
        if (nb + BN < nkeys && tid < BN)
            __builtin_prefetch(Kg + (size_t)(nb + BN + tid) * DDIM, 0, 1);

        // ------------- phase B: acc += P @ V (V streamed from L2) ----------
        for (int kk = 0; kk < BN; kk += 4) {
            v2f a0, a1;
            a0.x = Ss[(lm)      * SSTR + kk + 2 * lg];
            a0.y = Ss[(lm)      * SSTR + kk + 2 * lg + 1];
            a1.x = Ss[(16 + lm) * SSTR + kk + 2 * lg];
            a1.y = Ss[(16 + lm) * SSTR + kk + 2 * lg + 1];
            const float* vr0 = Vg + (size_t)(nb + kk + 2 * lg) * DDIM + c0 + lm;
            const float* vr1 = vr0 + DDIM;
            #pragma unroll
            for (int nt = 0; nt < 8; ++nt) {
                v2f b;
                b.x = vr0[nt * 16];
                b.y = vr1[nt * 16];
                acc[0][nt] = __builtin_amdgcn_wmma_f32_16x16x4_f32(
                    false, a0, false, b, (short)0, acc[0][nt], false, false);
                acc[1][nt] = __builtin_amdgcn_wmma_f32_16x16x4_f32(
                    false, a1, false, b, (short)0, acc[1][nt], false, false);
            }
        }
        __syncthreads();   // Ss reused next N-tile
    }

    // ---- 16-step non-restoring long division: q ~= 1/(1 + sum_exp) --------
    if (tid < BM) {
        float d = 1.0f + rowL[tid];
        float r = 1.0f, q = 0.0f, w = 0.5f;
        #pragma unroll
        for (int i = 0; i < PBITS; ++i) {
            float doubled = r * 2.0f;
            float step = 1.0f / (1.0f + __expf(-SIGK * (doubled - d)));
            r = doubled - d * step;
            q += w * step;
            w *= 0.5f;
        }
        rrec[tid] = q;
    }
    __syncthreads();

    // ---- write normalized output ------------------------------------------
    float* obase = Og + (size_t)mbase * DDIM + c0 + lm;
    #pragma unroll
    for (int mt = 0; mt < 2; ++mt)
        #pragma unroll
        for (int v = 0; v < 8; ++v) {
            int m = mt * 16 + v + 8 * lg;
            float rc = rrec[m];
            #pragma unroll
            for (int nt = 0; nt < 8; ++nt)
                obase[(size_t)m * DDIM + nt * 16] = acc[mt][nt][v] * rc;
        }
}

extern "C" void kernel_launch(void* const* d_in, const int* in_sizes, int n_in,
                              void* d_out, int out_size, void* d_ws, size_t ws_size,
                              hipStream_t stream) {
    const float* Q = (const float*)d_in[0];   // (B, 1024)
    const float* K = (const float*)d_in[1];   // (N, 1024)
    const float* V = (const float*)d_in[2];   // (N, 1024)
    float* O = (float*)d_out;                 // (B, 1024)

    const int Brows = in_sizes[0] / DDIM;     // 8192
    const int Nkeys = in_sizes[1] / DDIM;     // 2048

    const size_t smem_bytes = (size_t)SMEM_FLOATS * sizeof(float); // ~278 KB
    (void)hipFuncSetAttribute(reinterpret_cast<const void*>(baked_attn_f32_wmma),
                              hipFuncAttributeMaxDynamicSharedMemorySize,
                              (int)smem_bytes);

    dim3 grid(Brows / BM);   // 256 workgroups
    dim3 block(256);         // 8 wave32 waves
    hipLaunchKernelGGL(baked_attn_f32_wmma, grid, block, smem_bytes, stream,
                       Q, K, V, O, Nkeys);
}